// RoiPooling_11802570130056
// MI455X (gfx1250) — compile-verified
//
#include <hip/hip_runtime.h>
#include <hip/hip_bf16.h>

// ROI bilinear pooling reformulated per-ROI as out(49x256) = A(49x196) @ V(196x256)
// in f32 via V_WMMA_F32_16X16X4_F32, with the 14x14 source-pixel grid staged in
// CDNA5's 320KB-per-WGP LDS. One workgroup (8 wave32) per ROI.
//
// Round 2: stage V with GLOBAL_LOAD_ASYNC_TO_LDS_B128 (ASYNCcnt DMA path, no VGPR
// round-trip) and overlap it with the A-weight VALU phase; s_wait_asynccnt 0 before
// the barrier.

typedef float v2f __attribute__((ext_vector_type(2)));
typedef float v8f __attribute__((ext_vector_type(8)));

#define POOL    7
#define NP      49            // POOL*POOL output positions per ROI
#define GRID    14            // 2*POOL candidate rows / cols
#define NGRID   196           // GRID*GRID grid pixels (K dimension)
#define MPAD    64            // 49 padded to 4 WMMA M-tiles
#define CCH     256           // channels (N dimension)
#define IMW     200
#define VSTRIDE 272           // padded LDS row stride (floats) -> disjoint bank sets per half-wave

__global__ __launch_bounds__(256) void roi_pool_wmma(
    const float* __restrict__ img, const int* __restrict__ rois,
    float* __restrict__ out, int nroi)
{
  extern __shared__ float smem[];
  float* Vl     = smem;                                 // NGRID * VSTRIDE floats (213 KB), LDS offset 0
  float* Al     = smem + NGRID * VSTRIDE;               // MPAD * NGRID floats   (49 KB)
  int*   rowIdx = (int*)(Al + MPAD * NGRID);            // 14
  int*   colIdx = rowIdx + GRID;                        // 14
  float* fy     = (float*)(colIdx + GRID);              // 7
  float* fx     = fy + POOL;                            // 7

  const int tid = threadIdx.x;
  const int r   = blockIdx.x;
  if (r >= nroi) return;

  // ---- Phase 0: per-axis bilinear sample indices/weights (tf.resize half-pixel) ----
  if (tid < 2 * POOL) {
    const int axis  = tid / POOL;                 // 0 = y, 1 = x
    const int p     = tid % POOL;
    const int start = rois[r * 4 + (axis ? 0 : 1)];   // x for axis1, y for axis0
    const int size  = rois[r * 4 + (axis ? 2 : 3)];   // w for axis1, h for axis0
    const float scale = (float)size / (float)POOL;
    const float t   = ((float)p + 0.5f) * scale - 0.5f;
    const float fl  = floorf(t);
    const int   i0  = (int)fl;
    int lo = i0     < 0 ? 0 : (i0     > size - 1 ? size - 1 : i0);
    int hi = i0 + 1 < 0 ? 0 : (i0 + 1 > size - 1 ? size - 1 : i0 + 1);
    lo += start; hi += start;
    if (axis == 0) { rowIdx[p] = lo; rowIdx[p + POOL] = hi; fy[p] = t - fl; }
    else           { colIdx[p] = lo; colIdx[p + POOL] = hi; fx[p] = t - fl; }
  }
  __syncthreads();

  // ---- Phase 1: issue async DMA of V (196 grid pixels x 256 ch) into LDS ----
  // Per lane: 16B global -> 16B LDS, ASYNCcnt-tracked; overlaps with Phase 2 below.
  for (int t = tid; t < NGRID * 64; t += 256) {
    const int g = t >> 6, q = t & 63;
    const int row = rowIdx[g / GRID];
    const int col = colIdx[g % GRID];
    const float* gsrc = img + ((size_t)(row * IMW + col) * CCH + q * 4);
    const unsigned int ldsOff = (unsigned int)((g * VSTRIDE + q * 4) * sizeof(float));
    asm volatile("global_load_async_to_lds_b128 %0, %1, off"
                 :
                 : "v"(ldsOff), "v"((unsigned long long)(uintptr_t)gsrc)
                 : "memory");
  }

  // ---- Phase 2: build A (64x196 weight matrix) while the V DMA is in flight ----
  for (int t = tid; t < MPAD * NGRID; t += 256) {
    const int m = t / NGRID, k = t - m * NGRID;
    float wgt = 0.0f;
    if (m < NP) {
      const int py = m / POOL, px = m - py * POOL;
      const int gy = k / GRID, gx = k - gy * GRID;
      const float wy = (gy == py) ? (1.0f - fy[py]) : ((gy == py + POOL) ? fy[py] : 0.0f);
      const float wx = (gx == px) ? (1.0f - fx[px]) : ((gx == px + POOL) ? fx[px] : 0.0f);
      wgt = wy * wx;
    }
    Al[t] = wgt;
  }

  // All of this wave's async LDS writes must land before we signal the barrier.
  asm volatile("s_wait_asynccnt 0" ::: "memory");
  __syncthreads();

  // ---- Phase 3: out = A @ V with V_WMMA_F32_16X16X4_F32 ----
  // wave -> (1 M-tile) x (8 N-tiles); A fragment reused across the 8 WMMAs per k-step.
  const int lane   = tid & 31;
  const int wv     = tid >> 5;
  const int mtile  = wv & 3;             // 0..3  (M tiles of 16)
  const int ntbase = (wv >> 2) * 8;      // 0 or 8 (first of 8 N tiles)
  const int mrow   = mtile * 16 + (lane & 15);
  const int kofs   = (lane < 16) ? 0 : 2;   // A/B VGPR layout: halves hold K {0,1}/{2,3}

  v8f acc[8];
  const v8f zero = {0.f, 0.f, 0.f, 0.f, 0.f, 0.f, 0.f, 0.f};
#pragma unroll
  for (int t = 0; t < 8; ++t) acc[t] = zero;

  for (int k = 0; k < NGRID / 4; ++k) {
    const float* ap = Al + mrow * NGRID + k * 4 + kofs;   // 8B-aligned pair -> ds_load_b64
    v2f a; a.x = ap[0]; a.y = ap[1];
    const int g = k * 4 + kofs;
    const float* b0 = Vl + (size_t)g * VSTRIDE;
    const float* b1 = b0 + VSTRIDE;
#pragma unroll
    for (int t = 0; t < 8; ++t) {
      const int n = (ntbase + t) * 16 + (lane & 15);
      v2f b; b.x = b0[n]; b.y = b1[n];
      acc[t] = __builtin_amdgcn_wmma_f32_16x16x4_f32(
          /*neg_a=*/false, a, /*neg_b=*/false, b,
          /*c_mod=*/(short)0, acc[t], /*reuse_a=*/false, /*reuse_b=*/false);
    }
  }

  // ---- Phase 4: scatter D fragments (16x16 f32 C/D layout) to out[r,p,c] ----
  float* orow = out + (size_t)r * (NP * CCH);
  const int mbase = mtile * 16 + ((lane >= 16) ? 8 : 0);
#pragma unroll
  for (int t = 0; t < 8; ++t) {
    const int n = (ntbase + t) * 16 + (lane & 15);
#pragma unroll
    for (int v = 0; v < 8; ++v) {
      const int p = mbase + v;
      if (p < NP) orow[(size_t)p * CCH + n] = acc[t][v];
    }
  }
}

extern "C" void kernel_launch(void* const* d_in, const int* in_sizes, int n_in,
                              void* d_out, int out_size, void* d_ws, size_t ws_size,
                              hipStream_t stream) {
  const float* img  = (const float*)d_in[0];
  const int*   rois = (const int*)d_in[1];
  float*       out  = (float*)d_out;
  const int nroi = in_sizes[1] / 4;   // rois flat count / 4

  // 196*272 (V) + 64*196 (A) + 28 ints + 14 floats of axis data
  const size_t smem_bytes =
      (size_t)(NGRID * VSTRIDE + MPAD * NGRID + 2 * GRID + 2 * POOL) * sizeof(float);

  roi_pool_wmma<<<nroi, 256, smem_bytes, stream>>>(img, rois, out, nroi);
}